// SparseMoE_45844480917678
// MI455X (gfx1250) — compile-verified
//
#include <hip/hip_runtime.h>
#include <math.h>

// Problem constants (from the reference): B=2, S=2048 -> T=4096, D=1024, H=2048, E=8, top-2.
#define T_TOK 4096
#define DIM   1024
#define HID   2048
#define NEXP  8

typedef __attribute__((ext_vector_type(2))) float v2f;
typedef __attribute__((ext_vector_type(8))) float v8f;

// LDS strides:
//  A tile: 128 x 16, stride 18 -> rows 8B-aligned at even k, banks 18m%64 distinct for m<16.
//  B tile: 16 x 64 stored K-pair interleaved: [kpair(8)][n(64)][2], row stride 132 words.
#define LDA 18
#define LDB 132

// ---------------------------------------------------------------------------
// Kernel 1: router. One wave32 per token: 8 logits via lane-strided dot,
// cross-lane reduce, top-2 + softmax, atomic per-expert counts.
// ---------------------------------------------------------------------------
__global__ __launch_bounds__(256)
void moe_router_kernel(const float* __restrict__ x,
                       const float* __restrict__ rw,
                       int* __restrict__ cnt,
                       int* __restrict__ te,
                       float* __restrict__ tw) {
  const int tid  = threadIdx.x;
  const int wid  = tid >> 5;
  const int lane = tid & 31;
  const int t    = blockIdx.x * 8 + wid;

  float acc[NEXP];
#pragma unroll
  for (int e = 0; e < NEXP; ++e) acc[e] = 0.0f;

  const float* xr = x + (size_t)t * DIM;
  for (int d = lane; d < DIM; d += 32) {
    const float xv = xr[d];
    const float* r = rw + d * NEXP;
#pragma unroll
    for (int e = 0; e < NEXP; ++e) acc[e] = fmaf(xv, r[e], acc[e]);
  }
#pragma unroll
  for (int e = 0; e < NEXP; ++e) {
#pragma unroll
    for (int off = 16; off >= 1; off >>= 1)
      acc[e] += __shfl_xor(acc[e], off, 32);
  }
  if (lane == 0) {
    int e0 = 0; float l0 = acc[0];
#pragma unroll
    for (int e = 1; e < NEXP; ++e) if (acc[e] > l0) { l0 = acc[e]; e0 = e; }
    int e1 = -1; float l1 = -INFINITY;
#pragma unroll
    for (int e = 0; e < NEXP; ++e) if (e != e0 && acc[e] > l1) { l1 = acc[e]; e1 = e; }
    // softmax over [l0, l1] (l0 >= l1)
    const float w0 = 1.0f / (1.0f + __expf(l1 - l0));
    const float w1 = 1.0f - w0;
    te[2 * t]     = e0;
    te[2 * t + 1] = e1;
    tw[2 * t]     = w0;
    tw[2 * t + 1] = w1;
    atomicAdd(&cnt[e0], 1);
    atomicAdd(&cnt[e1], 1);
  }
}

// ---------------------------------------------------------------------------
// Kernel 2: exclusive prefix sum over 8 counts -> compact row bases.
// ---------------------------------------------------------------------------
__global__ void moe_base_kernel(const int* __restrict__ cnt,
                                int* __restrict__ base,
                                int* __restrict__ cursor) {
  if (threadIdx.x == 0 && blockIdx.x == 0) {
    int b = 0;
#pragma unroll
    for (int e = 0; e < NEXP; ++e) { base[e] = b; b += cnt[e]; cursor[e] = 0; }
  }
}

// ---------------------------------------------------------------------------
// Kernel 3: scatter tokens into per-expert compacted row lists.
// ---------------------------------------------------------------------------
__global__ __launch_bounds__(256)
void moe_place_kernel(const int* __restrict__ te,
                      const float* __restrict__ tw,
                      const int* __restrict__ base,
                      int* __restrict__ cursor,
                      int* __restrict__ rowtok,
                      float* __restrict__ roww) {
  const int t = blockIdx.x * blockDim.x + threadIdx.x;
  if (t >= T_TOK) return;
#pragma unroll
  for (int k = 0; k < 2; ++k) {
    const int e = te[2 * t + k];
    const int p = atomicAdd(&cursor[e], 1);
    const int g = base[e] + p;
    rowtok[g] = t;
    roww[g]   = tw[2 * t + k];
  }
}

// ---------------------------------------------------------------------------
// Kernel 4: grouped GEMM for gate+up, fused SiLU(gate)*up -> fused scratch.
// Block tile M=128 x N=64, K-step 16, fp32 WMMA 16x16x4, depth-1 register
// prefetch so global loads overlap the WMMA block of the previous K chunk.
// Grid: (HID/64, 32 row tiles, NEXP). Uniform early-exit keeps EXEC all-1s.
// ---------------------------------------------------------------------------
__global__ __launch_bounds__(256)
void moe_stage_a_kernel(const float* __restrict__ x,
                        const float* __restrict__ wg,
                        const float* __restrict__ wu,
                        const int* __restrict__ cnt,
                        const int* __restrict__ base,
                        const int* __restrict__ rowtok,
                        float* __restrict__ fused) {
  const int e    = blockIdx.z;
  const int c    = cnt[e];
  const int row0 = blockIdx.y * 128;
  if (row0 >= c) return;            // uniform across the block
  const int n0    = blockIdx.x * 64;
  const int gbase = base[e];

  __shared__ __align__(16) float lA [128 * LDA];
  __shared__ __align__(16) float lBg[8 * LDB];
  __shared__ __align__(16) float lBu[8 * LDB];
  __shared__ int sTok[128];

  const int tid = threadIdx.x;
  if (tid < 128) {
    const int r = row0 + tid;
    sTok[tid] = rowtok[gbase + (r < c ? r : c - 1)];
  }
  __syncthreads();

  const int lane = tid & 31;
  const int wid  = tid >> 5;
  const int m0   = wid * 16;        // wave owns rows m0..m0+15 x cols n0..n0+63
  const int kh   = lane >> 4;       // lane half selects K pair (A and B layouts)
  const int ml   = lane & 15;

  v8f accG[4] = {};
  v8f accU[4] = {};

  const size_t DH  = (size_t)DIM * HID;

  // Per-thread load slots.
  const int ar = tid >> 1;          // A: row 0..127
  const int ac = (tid & 1) * 8;     // A: col 0 or 8
  const int bk = tid >> 4;          // B: k row 0..15
  const int bn = (tid * 4) & 63;    // B: n offset (float4)

  const float* xrow = x  + (size_t)sTok[ar] * DIM + ac;
  const float* gptr = wg + (size_t)e * DH + (size_t)bk * HID + n0 + bn;
  const float* uptr = wu + (size_t)e * DH + (size_t)bk * HID + n0 + bn;

  float* sa  = &lA [ar * LDA + ac];
  float* sbg = &lBg[(bk >> 1) * LDB + bn * 2 + (bk & 1)];
  float* sbu = &lBu[(bk >> 1) * LDB + bn * 2 + (bk & 1)];

  // Prefetch chunk k0 = 0.
  float4 pa0 = *(const float4*)(xrow);
  float4 pa1 = *(const float4*)(xrow + 4);
  float4 pg  = *(const float4*)(gptr);
  float4 pu  = *(const float4*)(uptr);

  for (int k0 = 0; k0 < DIM; k0 += 16) {
    __syncthreads();                      // previous compute done -> LDS reusable
    sa[0] = pa0.x; sa[1] = pa0.y; sa[2] = pa0.z; sa[3] = pa0.w;
    sa[4] = pa1.x; sa[5] = pa1.y; sa[6] = pa1.z; sa[7] = pa1.w;
    sbg[0] = pg.x; sbg[2] = pg.y; sbg[4] = pg.z; sbg[6] = pg.w;
    sbu[0] = pu.x; sbu[2] = pu.y; sbu[4] = pu.z; sbu[6] = pu.w;
    __syncthreads();

    if (k0 + 16 < DIM) {                  // prefetch next chunk during compute
      pa0 = *(const float4*)(xrow + k0 + 16);
      pa1 = *(const float4*)(xrow + k0 + 20);
      pg  = *(const float4*)(gptr + (size_t)(k0 + 16) * HID);
      pu  = *(const float4*)(uptr + (size_t)(k0 + 16) * HID);
    }

#pragma unroll
    for (int ks = 0; ks < 4; ++ks) {
      const int kk = ks * 4 + 2 * kh;     // this lane's K pair within chunk
      const v2f a = *(const v2f*)&lA[(m0 + ml) * LDA + kk];
#pragma unroll
      for (int ns = 0; ns < 4; ++ns) {
        const int nn = ns * 16 + ml;
        const v2f bg = *(const v2f*)&lBg[(ks * 2 + kh) * LDB + nn * 2];
        const v2f bu = *(const v2f*)&lBu[(ks * 2 + kh) * LDB + nn * 2];
        accG[ns] = __builtin_amdgcn_wmma_f32_16x16x4_f32(
            false, a, false, bg, (short)0, accG[ns], false, false);
        accU[ns] = __builtin_amdgcn_wmma_f32_16x16x4_f32(
            false, a, false, bu, (short)0, accU[ns], false, false);
      }
    }
  }

  // Epilogue: fused = silu(gate) * up, write to compact fused scratch.
#pragma unroll
  for (int ns = 0; ns < 4; ++ns) {
#pragma unroll
    for (int v = 0; v < 8; ++v) {
      const int mc = v + 8 * kh;          // C/D layout: VGPR v -> M = v (+8 for hi half)
      const int r  = row0 + m0 + mc;
      if (r < c) {
        const float g = accG[ns][v];
        const float u = accU[ns][v];
        const float s = g / (1.0f + __expf(-g));
        fused[(size_t)(gbase + r) * HID + n0 + ns * 16 + ml] = s * u;
      }
    }
  }
}

// ---------------------------------------------------------------------------
// Kernel 5: grouped GEMM down-projection (K=2048), scale by routing weight,
// atomicAdd into y. Grid: (DIM/64, 32 row tiles, NEXP).
// ---------------------------------------------------------------------------
__global__ __launch_bounds__(256)
void moe_stage_b_kernel(const float* __restrict__ fused,
                        const float* __restrict__ wd,
                        const int* __restrict__ cnt,
                        const int* __restrict__ base,
                        const int* __restrict__ rowtok,
                        const float* __restrict__ roww,
                        float* __restrict__ y) {
  const int e    = blockIdx.z;
  const int c    = cnt[e];
  const int row0 = blockIdx.y * 128;
  if (row0 >= c) return;
  const int n0    = blockIdx.x * 64;
  const int gbase = base[e];

  __shared__ __align__(16) float lA[128 * LDA];
  __shared__ __align__(16) float lB[8 * LDB];
  __shared__ int   sTok[128];
  __shared__ float sW[128];

  const int tid = threadIdx.x;
  if (tid < 128) {
    const int r  = row0 + tid;
    const int rc = (r < c ? r : c - 1);
    sTok[tid] = rowtok[gbase + rc];
    sW[tid]   = roww[gbase + rc];
  }
  __syncthreads();

  const int lane = tid & 31;
  const int wid  = tid >> 5;
  const int m0   = wid * 16;
  const int kh   = lane >> 4;
  const int ml   = lane & 15;

  v8f acc[4] = {};

  const int ar = tid >> 1;
  const int ac = (tid & 1) * 8;
  const int bk = tid >> 4;
  const int bn = (tid * 4) & 63;

  // A rows come from the compact fused scratch (clamped rows are valid rows).
  const int arow = row0 + ar;
  const float* arp  = fused + (size_t)(gbase + (arow < c ? arow : c - 1)) * HID + ac;
  const float* bptr = wd + (size_t)e * HID * DIM + (size_t)bk * DIM + n0 + bn;

  float* sa = &lA[ar * LDA + ac];
  float* sb = &lB[(bk >> 1) * LDB + bn * 2 + (bk & 1)];

  float4 pa0 = *(const float4*)(arp);
  float4 pa1 = *(const float4*)(arp + 4);
  float4 pb  = *(const float4*)(bptr);

  for (int k0 = 0; k0 < HID; k0 += 16) {
    __syncthreads();
    sa[0] = pa0.x; sa[1] = pa0.y; sa[2] = pa0.z; sa[3] = pa0.w;
    sa[4] = pa1.x; sa[5] = pa1.y; sa[6] = pa1.z; sa[7] = pa1.w;
    sb[0] = pb.x;  sb[2] = pb.y;  sb[4] = pb.z;  sb[6] = pb.w;
    __syncthreads();

    if (k0 + 16 < HID) {
      pa0 = *(const float4*)(arp + k0 + 16);
      pa1 = *(const float4*)(arp + k0 + 20);
      pb  = *(const float4*)(bptr + (size_t)(k0 + 16) * DIM);
    }

#pragma unroll
    for (int ks = 0; ks < 4; ++ks) {
      const int kk = ks * 4 + 2 * kh;
      const v2f a = *(const v2f*)&lA[(m0 + ml) * LDA + kk];
#pragma unroll
      for (int ns = 0; ns < 4; ++ns) {
        const int nn = ns * 16 + ml;
        const v2f b = *(const v2f*)&lB[(ks * 2 + kh) * LDB + nn * 2];
        acc[ns] = __builtin_amdgcn_wmma_f32_16x16x4_f32(
            false, a, false, b, (short)0, acc[ns], false, false);
      }
    }
  }

  // Epilogue: y[token] += combine_weight * out_row
#pragma unroll
  for (int ns = 0; ns < 4; ++ns) {
#pragma unroll
    for (int v = 0; v < 8; ++v) {
      const int mc = v + 8 * kh;
      const int r  = row0 + m0 + mc;
      if (r < c) {
        const int   tok = sTok[m0 + mc];
        const float w   = sW[m0 + mc];
        atomicAdd(&y[(size_t)tok * DIM + n0 + ns * 16 + ml], w * acc[ns][v]);
      }
    }
  }
}

// ---------------------------------------------------------------------------
// Launch: router -> prefix -> place -> gate/up+SiLU -> down+combine.
// Workspace layout (bytes):
//   [0,32)        counts (8 int)
//   [32,64)       base   (8 int)
//   [64,96)       cursor (8 int)
//   [128, +32KB)  te (2T int)
//   ...           tw (2T f32), rowtok (2T int), roww (2T f32)
//   [131328, +64MB) fused scratch (8192 x 2048 f32)
// ---------------------------------------------------------------------------
extern "C" void kernel_launch(void* const* d_in, const int* in_sizes, int n_in,
                              void* d_out, int out_size, void* d_ws, size_t ws_size,
                              hipStream_t stream) {
  const float* x  = (const float*)d_in[0];
  const float* rw = (const float*)d_in[1];
  const float* wg = (const float*)d_in[2];
  const float* wu = (const float*)d_in[3];
  const float* wd = (const float*)d_in[4];
  float* y = (float*)d_out;

  char* ws = (char*)d_ws;
  int*   cnt    = (int*)(ws + 0);
  int*   base   = (int*)(ws + 32);
  int*   cursor = (int*)(ws + 64);
  int*   te     = (int*)(ws + 128);
  float* tw     = (float*)(ws + 128 + 1 * 32768);
  int*   rowtok = (int*)(ws + 128 + 2 * 32768);
  float* roww   = (float*)(ws + 128 + 3 * 32768);
  float* fused  = (float*)(ws + 131328);

  hipMemsetAsync(d_out, 0, (size_t)out_size * sizeof(float), stream);
  hipMemsetAsync(cnt, 0, 32, stream);

  moe_router_kernel<<<T_TOK / 8, 256, 0, stream>>>(x, rw, cnt, te, tw);
  moe_base_kernel<<<1, 32, 0, stream>>>(cnt, base, cursor);
  moe_place_kernel<<<T_TOK / 256, 256, 0, stream>>>(te, tw, base, cursor, rowtok, roww);
  moe_stage_a_kernel<<<dim3(HID / 64, T_TOK / 128, NEXP), 256, 0, stream>>>(
      x, wg, wu, cnt, base, rowtok, fused);
  moe_stage_b_kernel<<<dim3(DIM / 64, T_TOK / 128, NEXP), 256, 0, stream>>>(
      fused, wd, cnt, base, rowtok, roww, y);
}